// SpectralLinear_14293651161585
// MI455X (gfx1250) — compile-verified
//
#include <hip/hip_runtime.h>
#include <math.h>

typedef float v8f __attribute__((ext_vector_type(8)));
typedef float v2f __attribute__((ext_vector_type(2)));

#define INSIZE   1024
#define NH       20          // Householder reflectors per side
#define RPAD     1028        // LDS row stride (pad 4 -> conflict-free b64 A loads)
#define ROWS_WG  64
#define WAVES_WG 4
#define THREADS_WG 128

// workspace layout (float offsets). Pair-interleaved B operands: [k/2][col][2]
#define WS_YU    0                 // 32768: Yu packed  ([512][32][2]) Yu[k][c]=U[c][k]
#define WS_YVS   32768             // 32768: diag(s)*Yv packed
#define WS_UPK   65536             // 20480: U rows 0..19 packed ([10][1024][2])
#define WS_VPK   86016             // 20480: V rows 19..0 packed ([10][1024][2])
#define WS_GU    106496            // 400
#define WS_GV    106896            // 400
#define WS_TU    107296            // 400
#define WS_TV    107696            // 400
#define WS_S     108096            // 1024

__device__ __forceinline__ v8f wmma_f32_k4(v2f a, v2f b, v8f c) {
  // D(16x16,f32) = A(16x4,f32) * B(4x16,f32) + C
  return __builtin_amdgcn_wmma_f32_16x16x4_f32(false, a, false, b, (short)0, c,
                                               false, false);
}

// ---------------- prep 1: Gram matrices --------------------------------------
__global__ void prep_dots(const float* __restrict__ U, const float* __restrict__ V,
                          float* __restrict__ ws) {
  __shared__ float red[256];
  int b = blockIdx.x;               // 0..799
  int side = (b >= 400) ? 1 : 0;
  int idx = b - side * 400;
  int i = idx / NH, j = idx % NH;
  const float *ra, *rb;
  if (!side) { ra = U + i * INSIZE;        rb = U + j * INSIZE; }
  else       { ra = V + (19 - i) * INSIZE; rb = V + (19 - j) * INSIZE; }
  float s = 0.f;
  for (int k = threadIdx.x; k < INSIZE; k += 256) s += ra[k] * rb[k];
  red[threadIdx.x] = s;
  __syncthreads();
  for (int off = 128; off > 0; off >>= 1) {
    if ((int)threadIdx.x < off) red[threadIdx.x] += red[threadIdx.x + off];
    __syncthreads();
  }
  if (threadIdx.x == 0) ws[(side ? WS_GV : WS_GU) + i * NH + j] = red[0];
}

// ---------------- prep 2: sigmas + T matrices --------------------------------
__device__ void build_T(const float* __restrict__ G, float* __restrict__ T) {
  float tau[NH];
  for (int i = 0; i < NH; ++i) tau[i] = 2.0f / G[i * NH + i];
  for (int i = 0; i < NH * NH; ++i) T[i] = 0.f;
  for (int i = 0; i < NH; ++i) T[i * NH + i] = tau[i];
  // forward product H_1 H_2 ... H_k = I - Y T Y^T  (LAPACK larft recursion)
  for (int j = 1; j < NH; ++j)
    for (int i = 0; i < j; ++i) {
      float t = 0.f;
      for (int mm = i; mm < j; ++mm) t += T[i * NH + mm] * G[mm * NH + j];
      T[i * NH + j] = -tau[j] * t;
    }
}

__global__ void prep_build(const float* __restrict__ p, float* __restrict__ ws) {
  int tid = threadIdx.x;
  for (int k = tid; k < INSIZE; k += blockDim.x) {
    float sig = 1.f / (1.f + __expf(-p[k]));
    ws[WS_S + k] = 0.4f * (sig - 0.5f) + 0.8f;   // 2*R*(sig-0.5)+SIGMA_MEAN
  }
  if (tid == 0) build_T(ws + WS_GU, ws + WS_TU);
  else if (tid == 1) build_T(ws + WS_GV, ws + WS_TV);
}

// ---------------- prep 3: pack Y_U and diag(s)*Y_V (pair-interleaved) --------
__global__ void prep_fill(const float* __restrict__ U, const float* __restrict__ V,
                          float* __restrict__ ws) {
  int e = blockIdx.x * blockDim.x + threadIdx.x;   // 32768 elements
  int c = e >> 10, k = e & 1023;                   // coalesced over k
  float yu = 0.f, yv = 0.f;
  if (c < NH) {
    yu = U[c * INSIZE + k];
    yv = ws[WS_S + k] * V[(19 - c) * INSIZE + k];
  }
  int pk = ((k >> 1) << 6) + (c << 1) + (k & 1);   // [k/2][c][2], 32 cols
  ws[WS_YU  + pk] = yu;
  ws[WS_YVS + pk] = yv;
}

// ---------------- prep 4: pack U rows / reversed V rows ----------------------
__global__ void prep_pack(const float* __restrict__ U, const float* __restrict__ V,
                          float* __restrict__ ws) {
  int e = blockIdx.x * blockDim.x + threadIdx.x;   // 20480 elements
  int k = e >> 10, n = e & 1023;                   // k in 0..19, coalesced over n
  int pk = (k >> 1) * 2048 + (n << 1) + (k & 1);   // [k/2][n][2]
  ws[WS_UPK + pk] = U[k * INSIZE + n];
  ws[WS_VPK + pk] = V[(19 - k) * INSIZE + n];      // YvT row k = V[19-k]
}

// ---------------- main fused kernel ------------------------------------------
__global__ __launch_bounds__(THREADS_WG, 1)
void spectral_main(const float* __restrict__ x, const float* __restrict__ bias,
                   const float* __restrict__ ws, float* __restrict__ out) {
  __shared__ float rows[ROWS_WG * RPAD];            // 64 x 1024 row tile (padded)
  __shared__ float zbuf[WAVES_WG][16][32];
  __shared__ float wbuf[WAVES_WG][16][32];
  __shared__ float TuL[NH * NH], TvL[NH * NH];
  __shared__ float sL[INSIZE], biasL[INSIZE];

  const int tid  = threadIdx.x;
  const int lane = tid & 31, wid = tid >> 5;
  const int row0 = blockIdx.x * ROWS_WG;

  // ---- P0: async global->LDS stage of the X tile (ASYNCcnt path) ----
  {
    for (int idx = tid; idx < ROWS_WG * (INSIZE / 16); idx += THREADS_WG) {
      int r = idx >> 6, c16 = idx & 63;             // 64 x 16B-chunks per row
      unsigned lds_addr = (unsigned)(unsigned long long)&rows[r * RPAD + c16 * 4];
      unsigned goff = ((unsigned)(row0 + r) * INSIZE + c16 * 4) * 4u;
      asm volatile("global_load_async_to_lds_b128 %0, %1, %2"
                   :: "v"(lds_addr), "v"(goff), "s"(x) : "memory");
    }
    for (int i = tid; i < NH * NH; i += THREADS_WG) {
      TuL[i] = ws[WS_TU + i];
      TvL[i] = ws[WS_TV + i];
    }
    for (int i = tid; i < INSIZE; i += THREADS_WG) {
      sL[i] = ws[WS_S + i];
      biasL[i] = bias[i];
    }
    asm volatile("s_wait_asynccnt 0x0" ::: "memory");
  }
  __syncthreads();

  const int m    = lane & 15;              // A-fragment row (ISA 16x4 f32 layout)
  const int koff = (lane < 16) ? 0 : 2;    // lanes 16-31 carry K+2,K+3
  const int kph  = koff >> 1;
  const int nloc = lane & 15;              // B/C fragment column
  const int R0   = wid * 16;               // this wave's 16 rows

  // ---- P1: z = Xtile @ Yu   (16x32 per wave, K=1024, split accumulators) ----
  {
    const float* YuP = ws + WS_YU;
    v8f a0e = {0,0,0,0,0,0,0,0}, a0o = {0,0,0,0,0,0,0,0};
    v8f a1e = {0,0,0,0,0,0,0,0}, a1o = {0,0,0,0,0,0,0,0};
    const float* arow = &rows[(R0 + m) * RPAD];
#pragma unroll 2
    for (int kb2 = 0; kb2 < 128; ++kb2) {
      int ke = kb2 * 8 + koff;             // even kb
      int pe = kb2 * 4 + kph;              // pair index for even kb
      v2f ae = *(const v2f*)(arow + ke);
      v2f ao = *(const v2f*)(arow + ke + 4);
      v2f be0 = *(const v2f*)&YuP[pe * 64 + nloc * 2];
      v2f be1 = *(const v2f*)&YuP[pe * 64 + 32 + nloc * 2];
      v2f bo0 = *(const v2f*)&YuP[(pe + 2) * 64 + nloc * 2];
      v2f bo1 = *(const v2f*)&YuP[(pe + 2) * 64 + 32 + nloc * 2];
      a0e = wmma_f32_k4(ae, be0, a0e);
      a1e = wmma_f32_k4(ae, be1, a1e);
      a0o = wmma_f32_k4(ao, bo0, a0o);
      a1o = wmma_f32_k4(ao, bo1, a1o);
    }
    v8f acc0 = a0e + a0o, acc1 = a1e + a1o;
#pragma unroll
    for (int g = 0; g < 8; ++g) {
      int mm = (lane < 16) ? g : (g + 8);
      zbuf[wid][mm][nloc]      = acc0[g];
      zbuf[wid][mm][16 + nloc] = acc1[g];
    }
  }
  __syncthreads();

  // ---- w = -(z @ T_U)  (tiny, VALU) ----
  for (int idx = lane; idx < 16 * NH; idx += 32) {
    int mm = idx / NH, j = idx % NH;
    float s = 0.f;
    for (int i = 0; i <= j; ++i) s += zbuf[wid][mm][i] * TuL[i * NH + j];
    wbuf[wid][mm][j] = -s;
  }
  __syncthreads();

  // ---- P2: rows += (-w) @ U[0:20,:] ----
  {
    const float* Upk = ws + WS_UPK;
    v2f afr[5];
#pragma unroll
    for (int kb = 0; kb < 5; ++kb) {
      int k = kb * 4 + koff;
      afr[kb].x = wbuf[wid][m][k];
      afr[kb].y = wbuf[wid][m][k + 1];
    }
    const int mbase = R0 + ((lane < 16) ? 0 : 8);
#pragma unroll 2
    for (int nt = 0; nt < 64; ++nt) {
      int n = nt * 16 + nloc;
      v8f c;
#pragma unroll
      for (int g = 0; g < 8; ++g) c[g] = rows[(mbase + g) * RPAD + n];
#pragma unroll
      for (int kb = 0; kb < 5; ++kb) {
        v2f b = *(const v2f*)&Upk[(kb * 2 + kph) * 2048 + n * 2];
        c = wmma_f32_k4(afr[kb], b, c);
      }
#pragma unroll
      for (int g = 0; g < 8; ++g) rows[(mbase + g) * RPAD + n] = c[g];
    }
  }
  __syncthreads();

  // ---- P3: z2 = r1 @ (diag(s) Yv) ----
  {
    const float* YvP = ws + WS_YVS;
    v8f a0e = {0,0,0,0,0,0,0,0}, a0o = {0,0,0,0,0,0,0,0};
    v8f a1e = {0,0,0,0,0,0,0,0}, a1o = {0,0,0,0,0,0,0,0};
    const float* arow = &rows[(R0 + m) * RPAD];
#pragma unroll 2
    for (int kb2 = 0; kb2 < 128; ++kb2) {
      int ke = kb2 * 8 + koff;
      int pe = kb2 * 4 + kph;
      v2f ae = *(const v2f*)(arow + ke);
      v2f ao = *(const v2f*)(arow + ke + 4);
      v2f be0 = *(const v2f*)&YvP[pe * 64 + nloc * 2];
      v2f be1 = *(const v2f*)&YvP[pe * 64 + 32 + nloc * 2];
      v2f bo0 = *(const v2f*)&YvP[(pe + 2) * 64 + nloc * 2];
      v2f bo1 = *(const v2f*)&YvP[(pe + 2) * 64 + 32 + nloc * 2];
      a0e = wmma_f32_k4(ae, be0, a0e);
      a1e = wmma_f32_k4(ae, be1, a1e);
      a0o = wmma_f32_k4(ao, bo0, a0o);
      a1o = wmma_f32_k4(ao, bo1, a1o);
    }
    v8f acc0 = a0e + a0o, acc1 = a1e + a1o;
#pragma unroll
    for (int g = 0; g < 8; ++g) {
      int mm = (lane < 16) ? g : (g + 8);
      zbuf[wid][mm][nloc]      = acc0[g];
      zbuf[wid][mm][16 + nloc] = acc1[g];
    }
  }
  __syncthreads();

  // ---- w2 = -(z2 @ T_V) ----
  for (int idx = lane; idx < 16 * NH; idx += 32) {
    int mm = idx / NH, j = idx % NH;
    float s = 0.f;
    for (int i = 0; i <= j; ++i) s += zbuf[wid][mm][i] * TvL[i * NH + j];
    wbuf[wid][mm][j] = -s;
  }
  __syncthreads();

  // ---- P4: out = s*r1 + (-w2) @ YvT + bias ----
  {
    const float* Vpk = ws + WS_VPK;
    v2f afr[5];
#pragma unroll
    for (int kb = 0; kb < 5; ++kb) {
      int k = kb * 4 + koff;
      afr[kb].x = wbuf[wid][m][k];
      afr[kb].y = wbuf[wid][m][k + 1];
    }
    const int mbase = R0 + ((lane < 16) ? 0 : 8);
#pragma unroll 2
    for (int nt = 0; nt < 64; ++nt) {
      int n = nt * 16 + nloc;
      v8f c = {0, 0, 0, 0, 0, 0, 0, 0};
#pragma unroll
      for (int kb = 0; kb < 5; ++kb) {
        v2f b = *(const v2f*)&Vpk[(kb * 2 + kph) * 2048 + n * 2];
        c = wmma_f32_k4(afr[kb], b, c);
      }
#pragma unroll
      for (int g = 0; g < 8; ++g) {
        float r1v = rows[(mbase + g) * RPAD + n];
        out[(size_t)(row0 + mbase + g) * INSIZE + n] = sL[n] * r1v + c[g] + biasL[n];
      }
    }
  }
}

// ---------------- launcher ----------------------------------------------------
extern "C" void kernel_launch(void* const* d_in, const int* in_sizes, int n_in,
                              void* d_out, int out_size, void* d_ws, size_t ws_size,
                              hipStream_t stream) {
  const float* x    = (const float*)d_in[0];
  const float* U    = (const float*)d_in[1];
  const float* p    = (const float*)d_in[2];
  const float* V    = (const float*)d_in[3];
  const float* bias = (const float*)d_in[4];
  float* out = (float*)d_out;
  float* ws  = (float*)d_ws;
  int batch = in_sizes[0] / INSIZE;   // 32768

  prep_dots<<<2 * NH * NH, 256, 0, stream>>>(U, V, ws);
  prep_build<<<1, 256, 0, stream>>>(p, ws);
  prep_fill<<<(INSIZE * 32) / 256, 256, 0, stream>>>(U, V, ws);
  prep_pack<<<(NH * INSIZE) / 256, 256, 0, stream>>>(U, V, ws);
  spectral_main<<<batch / ROWS_WG, THREADS_WG, 0, stream>>>(x, bias, ws, out);
}